// HybridQuantumLoss_9053791060382
// MI455X (gfx1250) — compile-verified
//
#include <hip/hip_runtime.h>
#include <hip/hip_bf16.h>
#include <math.h>

#define NUM_Q 37
#define NUM_DRAWS 7
#define NUM_LAYERS 4
#define ALPHA_C 0.3f
#define BETA_C 0.2f
#define LS_C 0.1f

#define BPB 128      // batch rows per block
#define ST 41        // state LDS row stride (coprime with 64 banks)
#define PS 49        // p-buffer LDS row stride

// workspace float layout
#define WS_TRE 0          // [40][48] padded real total (K-major)
#define WS_TIM 1920       // [40][48] padded imag total
#define WS_U   3840       // u[40]
#define WS_ULOGU 3880     // scalar
#define WS_BASE 4096      // per-block partials (<=2048)
#define WS_KLP  6144
#define WS_DIV  8192

typedef __attribute__((ext_vector_type(2))) float v2f;
typedef __attribute__((ext_vector_type(8))) float v8f;

// ---------------------------------------------------------------------------
// Kernel 1: build total matrix (37x37 complex), opt_unif distribution u.
// ---------------------------------------------------------------------------
__global__ void k_total(const float* __restrict__ rot, const float* __restrict__ ent,
                        float* __restrict__ ws)
{
    __shared__ float cR[NUM_Q * 40], cI[NUM_Q * 40];
    __shared__ float tR[NUM_Q * 40], tI[NUM_Q * 40];
    __shared__ float eR[NUM_Q * NUM_Q], eI[NUM_Q * NUM_Q];
    __shared__ float pu[NUM_Q];

    const int tid = threadIdx.x;  // blockDim = 128

    for (int i = tid; i < NUM_Q * 40; i += 128) { cR[i] = 0.f; cI[i] = 0.f; }
    for (int i = tid; i < NUM_Q * NUM_Q; i += 128) {
        float e = ent[i];
        eR[i] = cosf(e); eI[i] = sinf(e);
    }
    __syncthreads();
    for (int i = tid; i < NUM_Q; i += 128) cR[i * 40 + i] = 1.0f;
    __syncthreads();

    for (int layer = 0; layer < NUM_LAYERS; ++layer) {
        // ACTIVE order: q ascending; only q=31..36 have pairs (mask = 1<<(36-q))
        for (int q = 31; q < NUM_Q; ++q) {
            const int mask = 1 << (NUM_Q - 1 - q);
            const float phi = rot[((layer * NUM_Q) + q) * 3 + 0];
            const float th  = rot[((layer * NUM_Q) + q) * 3 + 1];
            const float om  = rot[((layer * NUM_Q) + q) * 3 + 2];
            const float c = cosf(th * 0.5f), s = sinf(th * 0.5f);
            const float Aa = (phi + om) * 0.5f, Bb = (phi - om) * 0.5f;
            const float cA = cosf(Aa), sA = sinf(Aa), cB = cosf(Bb), sB = sinf(Bb);
            const float m00r =  c * cA, m00i = -c * sA;
            const float m01r = -s * cB, m01i = -s * sB;
            const float m10r =  s * cB, m10i = -s * sB;
            const float m11r =  c * cA, m11i =  c * sA;
            // total = total @ M  =>  column-pair updates, disjoint per (r,i)
            for (int idx = tid; idx < NUM_Q * NUM_Q; idx += 128) {
                const int r = idx / NUM_Q, i = idx % NUM_Q;
                const int t = i + mask;
                if (((i & mask) == 0) && t < NUM_Q) {
                    float ar = cR[r * 40 + i], ai = cI[r * 40 + i];
                    float br = cR[r * 40 + t], bi = cI[r * 40 + t];
                    cR[r * 40 + i] = m00r * ar - m00i * ai + m10r * br - m10i * bi;
                    cI[r * 40 + i] = m00r * ai + m00i * ar + m10r * bi + m10i * br;
                    cR[r * 40 + t] = m01r * ar - m01i * ai + m11r * br - m11i * bi;
                    cI[r * 40 + t] = m01r * ai + m01i * ar + m11r * bi + m11i * br;
                }
            }
            __syncthreads();
        }
        // total = total @ exp(i*E)   (full 37x37x37 complex matmul, tiny)
        for (int idx = tid; idx < NUM_Q * NUM_Q; idx += 128) {
            const int r = idx / NUM_Q, j = idx % NUM_Q;
            float accr = 0.f, acci = 0.f;
            for (int k = 0; k < NUM_Q; ++k) {
                float xr = cR[r * 40 + k], xi = cI[r * 40 + k];
                float er = eR[k * NUM_Q + j], ei = eI[k * NUM_Q + j];
                accr += xr * er - xi * ei;
                acci += xr * ei + xi * er;
            }
            tR[r * 40 + j] = accr; tI[r * 40 + j] = acci;
        }
        __syncthreads();
        for (int idx = tid; idx < NUM_Q * NUM_Q; idx += 128) {
            const int r = idx / NUM_Q, j = idx % NUM_Q;
            cR[r * 40 + j] = tR[r * 40 + j];
            cI[r * 40 + j] = tI[r * 40 + j];
        }
        __syncthreads();
    }

    // store padded K-major total: T[k][j] -> out[j] = sum_k state[k]*T[k][j]
    for (int i = tid; i < 40 * 48; i += 128) {
        const int k = i / 48, j = i % 48;
        float vr = 0.f, vi = 0.f;
        if (k < NUM_Q && j < NUM_Q) { vr = cR[k * 40 + j]; vi = cI[k * 40 + j]; }
        ws[WS_TRE + i] = vr;
        ws[WS_TIM + i] = vi;
    }

    // opt_unif: uniform probs -> state -> |state@total|^2 normalized
    const float sv = sqrtf(1.0f / NUM_Q + 1e-6f);
    const float nv = sqrtf((float)NUM_Q * sv * sv);
    const float su = sv / nv;
    if (tid < NUM_Q) {
        float accr = 0.f, acci = 0.f;
        for (int k = 0; k < NUM_Q; ++k) {
            accr += su * cR[k * 40 + tid];
            acci += su * cI[k * 40 + tid];
        }
        pu[tid] = accr * accr + acci * acci;
    }
    __syncthreads();
    if (tid == 0) {
        float ps = 0.f;
        for (int j = 0; j < NUM_Q; ++j) ps += pu[j];
        float ul = 0.f;
        for (int j = 0; j < NUM_Q; ++j) {
            float u = pu[j] / ps;
            ws[WS_U + j] = u;
            ul += u * logf(u);
        }
        for (int j = NUM_Q; j < 40; ++j) ws[WS_U + j] = 0.f;
        ws[WS_ULOGU] = ul;
    }
}

// ---------------------------------------------------------------------------
// Kernel 2: fused streaming pass over outputs. One thread per batch row for
// scalar phases; WMMA f32 16x16x4 for the (state @ total) complex matmul.
// ---------------------------------------------------------------------------
__global__ void k_main(const float* __restrict__ outputs, const int* __restrict__ targets,
                       const float* __restrict__ wsc, float* __restrict__ wsp, int Bn)
{
    __shared__ float sm[BPB * ST + BPB * PS + 2 * 40 * 48 + 40 + 128];
    float* stateS = sm;                       // [BPB][ST]
    float* pbufS  = sm + BPB * ST;            // [BPB][PS]; doubles as sbuf (stride ST) in phase A
    float* TReS   = pbufS + BPB * PS;         // [40][48]
    float* TImS   = TReS + 40 * 48;           // [40][48]
    float* uS     = TImS + 40 * 48;           // [40]
    float* redS   = uS + 40;                  // [128]

    const int tid = threadIdx.x;              // blockDim = 128 (4 waves)
    const int b   = blockIdx.x * BPB + tid;

    for (int i = tid; i < 40 * 48; i += 128) { TReS[i] = wsc[WS_TRE + i]; TImS[i] = wsc[WS_TIM + i]; }
    if (tid < 40) uS[tid] = wsc[WS_U + tid];
    for (int i = tid; i < BPB * ST; i += 128) stateS[i] = 0.f;
    for (int i = tid; i < BPB * PS; i += 128) pbufS[i] = 0.f;
    __syncthreads();

    // ---- Phase A: softmax stats, CE terms, div terms, mean-prob state ----
    float baseAcc = 0.f, divAcc = 0.f;
    if (b < Bn) {
        const float* rowp = outputs + (size_t)b * (NUM_DRAWS * NUM_Q);
        float* st = stateS + tid * ST;
        float* sb = pbufS + tid * ST;   // sum of normalized draws
        float divDiag = 0.f;
        for (int d = 0; d < NUM_DRAWS; ++d) {
            float x[NUM_Q];
            float mx = -3.402823466e38f, sx = 0.f, nsq = 0.f;
            #pragma unroll
            for (int c = 0; c < NUM_Q; ++c) {
                x[c] = rowp[d * NUM_Q + c];
                mx = fmaxf(mx, x[c]);
                sx += x[c];
                nsq += x[c] * x[c];
            }
            const float nrmv = sqrtf(nsq);
            const float ninv = 1.0f / fmaxf(nrmv, 1e-8f);
            const float nn = nrmv * ninv;
            divDiag += nn * nn;
            #pragma unroll
            for (int c = 0; c < NUM_Q; ++c) sb[c] += x[c] * ninv;

            const int tgt = targets[b * NUM_DRAWS + d];
            float se = 0.f, xt = x[0];
            #pragma unroll
            for (int c = 0; c < NUM_Q; ++c) {
                if (c == tgt) xt = x[c];
                float e = expf(x[c] - mx);
                x[c] = e;
                se += e;
            }
            const float lse = mx + logf(se);
            const float nll = lse - xt;
            const float smooth = lse - sx * (1.0f / NUM_Q);
            baseAcc += (1.0f - LS_C) * nll + LS_C * smooth;
            const float sei = 1.0f / se;
            #pragma unroll
            for (int c = 0; c < NUM_Q; ++c) st[c] += x[c] * sei;
        }
        // state = normalize(sqrt(mean_probs + 1e-6))
        float ns = 0.f;
        #pragma unroll
        for (int c = 0; c < NUM_Q; ++c) {
            float y = sqrtf(st[c] * (1.0f / NUM_DRAWS) + 1e-6f);
            st[c] = y;
            ns += y * y;
        }
        const float nsi = 1.0f / sqrtf(ns);
        #pragma unroll
        for (int c = 0; c < NUM_Q; ++c) st[c] *= nsi;
        // div raw: ||sum nrm||^2 - sum ||nrm||^2
        float ssq = 0.f;
        #pragma unroll
        for (int c = 0; c < NUM_Q; ++c) ssq += sb[c] * sb[c];
        divAcc = ssq - divDiag;
    }
    __syncthreads();

    // ---- Phase B: p = |state @ total|^2 via v_wmma_f32_16x16x4_f32 ----
    {
        const int wave = tid >> 5;
        const int lane = tid & 31;
        const int m    = lane & 15;
        const int kp   = (lane >> 4) * 2;     // A: VGPR0 holds K=0|K=2 halves, VGPR1 K=1|K=3
        const int ncol = lane & 15;
        for (int mb = wave * 16; mb < BPB; mb += 64) {
            v8f accRe[3], accIm[3];
            #pragma unroll
            for (int nt = 0; nt < 3; ++nt) {
                accRe[nt] = (v8f){0.f,0.f,0.f,0.f,0.f,0.f,0.f,0.f};
                accIm[nt] = (v8f){0.f,0.f,0.f,0.f,0.f,0.f,0.f,0.f};
            }
            const float* srow = stateS + (mb + m) * ST;
            for (int k0 = 0; k0 < 40; k0 += 4) {
                v2f a;
                a.x = srow[k0 + kp];
                a.y = srow[k0 + kp + 1];
                #pragma unroll
                for (int nt = 0; nt < 3; ++nt) {
                    const int col = nt * 16 + ncol;
                    v2f br, bi;
                    br.x = TReS[(k0 + kp) * 48 + col];
                    br.y = TReS[(k0 + kp + 1) * 48 + col];
                    bi.x = TImS[(k0 + kp) * 48 + col];
                    bi.y = TImS[(k0 + kp + 1) * 48 + col];
                    accRe[nt] = __builtin_amdgcn_wmma_f32_16x16x4_f32(
                        false, a, false, br, (short)0, accRe[nt], false, false);
                    accIm[nt] = __builtin_amdgcn_wmma_f32_16x16x4_f32(
                        false, a, false, bi, (short)0, accIm[nt], false, false);
                }
            }
            // D layout: VGPR r = row (r + (lane>=16 ? 8 : 0)), col = lane&15
            const int rbase = mb + (lane >> 4) * 8;
            #pragma unroll
            for (int nt = 0; nt < 3; ++nt) {
                const int col = nt * 16 + ncol;
                #pragma unroll
                for (int r = 0; r < 8; ++r) {
                    const float re = accRe[nt][r], im = accIm[nt][r];
                    pbufS[(rbase + r) * PS + col] = re * re + im * im;
                }
            }
        }
    }
    __syncthreads();

    // ---- Phase C: normalize p, KL inner product ----
    float klAcc = 0.f;
    if (b < Bn) {
        const float* pr = pbufS + tid * PS;
        float ps = 0.f;
        #pragma unroll
        for (int q = 0; q < NUM_Q; ++q) ps += pr[q];
        const float psi = 1.0f / ps;
        #pragma unroll
        for (int q = 0; q < NUM_Q; ++q) klAcc += uS[q] * logf(pr[q] * psi + 1e-6f);
    }

    // ---- deterministic block tree-reductions ----
    redS[tid] = baseAcc; __syncthreads();
    for (int s = 64; s > 0; s >>= 1) { if (tid < s) redS[tid] += redS[tid + s]; __syncthreads(); }
    if (tid == 0) wsp[WS_BASE + blockIdx.x] = redS[0];
    __syncthreads();

    redS[tid] = klAcc; __syncthreads();
    for (int s = 64; s > 0; s >>= 1) { if (tid < s) redS[tid] += redS[tid + s]; __syncthreads(); }
    if (tid == 0) wsp[WS_KLP + blockIdx.x] = redS[0];
    __syncthreads();

    redS[tid] = divAcc; __syncthreads();
    for (int s = 64; s > 0; s >>= 1) { if (tid < s) redS[tid] += redS[tid + s]; __syncthreads(); }
    if (tid == 0) wsp[WS_DIV + blockIdx.x] = redS[0];
}

// ---------------------------------------------------------------------------
// Kernel 3: deterministic final reduction + scalar combine.
// ---------------------------------------------------------------------------
__global__ void k_final(const float* __restrict__ ws, float* __restrict__ out,
                        int nblocks, int Bn)
{
    __shared__ float r0[256], r1[256], r2[256];
    const int tid = threadIdx.x;  // 256
    float bs = 0.f, ks = 0.f, ds = 0.f;
    for (int i = tid; i < nblocks; i += 256) {
        bs += ws[WS_BASE + i];
        ks += ws[WS_KLP + i];
        ds += ws[WS_DIV + i];
    }
    r0[tid] = bs; r1[tid] = ks; r2[tid] = ds;
    __syncthreads();
    for (int s = 128; s > 0; s >>= 1) {
        if (tid < s) { r0[tid] += r0[tid + s]; r1[tid] += r1[tid + s]; r2[tid] += r2[tid + s]; }
        __syncthreads();
    }
    if (tid == 0) {
        const float Bf = (float)Bn;
        const float base = r0[0] / (Bf * NUM_DRAWS);
        const float kl   = ws[WS_ULOGU] - r1[0] / Bf;
        const float dv   = r2[0] / (Bf * (float)(NUM_DRAWS * NUM_DRAWS));
        out[0] = base + ALPHA_C * kl + BETA_C * dv;
    }
}

// ---------------------------------------------------------------------------
extern "C" void kernel_launch(void* const* d_in, const int* in_sizes, int n_in,
                              void* d_out, int out_size, void* d_ws, size_t ws_size,
                              hipStream_t stream)
{
    const float* outputs = (const float*)d_in[0];
    const int*   targets = (const int*)d_in[1];
    const float* rot     = (const float*)d_in[2];
    const float* ent     = (const float*)d_in[3];
    float* ws  = (float*)d_ws;
    float* out = (float*)d_out;

    const int Bn = in_sizes[0] / (NUM_DRAWS * NUM_Q);
    const int nblocks = (Bn + BPB - 1) / BPB;

    k_total<<<1, 128, 0, stream>>>(rot, ent, ws);
    k_main<<<nblocks, 128, 0, stream>>>(outputs, targets, ws, ws, Bn);
    k_final<<<1, 256, 0, stream>>>(ws, out, nblocks, Bn);
}